// SPDeepGENLayer_19404662243618
// MI455X (gfx1250) — compile-verified
//
#include <hip/hip_runtime.h>
#include <math.h>

#define C   128
#define C2  256
#define NN  20000
#define EE  320000
#define LEE 640000
#define GEPS 1e-7f

typedef __attribute__((ext_vector_type(2))) float v2f;
typedef __attribute__((ext_vector_type(8))) float v8f;

#define WMMA_F32(a, b, c) \
    __builtin_amdgcn_wmma_f32_16x16x4_f32(false, (a), false, (b), (short)0, (c), false, false)

__device__ __forceinline__ float wave_sum(float v) {
    #pragma unroll
    for (int off = 16; off > 0; off >>= 1) v += __shfl_xor(v, off, 32);
    return v;
}

__device__ __forceinline__ void atomicMaxFloat(float* addr, float val) {
    if (val >= 0.0f) {
        atomicMax((int*)addr, __float_as_int(val));
    } else {
        atomicMin((unsigned int*)addr, __float_as_uint(val));
    }
}

// ---------------------------------------------------------------------------
// h = relu(LayerNorm(x))   one wave32 per row, 4 floats per lane
// ---------------------------------------------------------------------------
__global__ void prenorm_kernel(const float* __restrict__ x,
                               const float* __restrict__ g,
                               const float* __restrict__ b,
                               float* __restrict__ h, int nrows) {
    int w = threadIdx.x >> 5, lane = threadIdx.x & 31;
    int row = blockIdx.x * 8 + w;
    if (row >= nrows) return;
    float4 v = ((const float4*)(x + (size_t)row * C))[lane];
    float s  = v.x + v.y + v.z + v.w;
    float sq = v.x * v.x + v.y * v.y + v.z * v.z + v.w * v.w;
    s = wave_sum(s); sq = wave_sum(sq);
    float m  = s * (1.0f / C);
    float vr = sq * (1.0f / C) - m * m;
    float rs = rsqrtf(vr + 1e-5f);
    float4 gv = ((const float4*)g)[lane];
    float4 bv = ((const float4*)b)[lane];
    float4 o;
    o.x = fmaxf((v.x - m) * rs * gv.x + bv.x, 0.0f);
    o.y = fmaxf((v.y - m) * rs * gv.y + bv.y, 0.0f);
    o.z = fmaxf((v.z - m) * rs * gv.z + bv.z, 0.0f);
    o.w = fmaxf((v.w - m) * rs * gv.w + bv.w, 0.0f);
    ((float4*)(h + (size_t)row * C))[lane] = o;
}

__global__ void fill_kernel(float* __restrict__ p, float val, size_t n) {
    size_t i = (size_t)blockIdx.x * blockDim.x + threadIdx.x;
    size_t st = (size_t)gridDim.x * blockDim.x;
    for (; i < n; i += st) p[i] = val;
}

// ---------------------------------------------------------------------------
// Pass A: per-destination, per-channel segment max of logits
// ---------------------------------------------------------------------------
__global__ void scatter_max_kernel(const float* __restrict__ h,
                                   const float* __restrict__ eattr,
                                   const int* __restrict__ src,
                                   const int* __restrict__ dst,
                                   const float* __restrict__ tptr,
                                   float* mx, int nE) {
    int w = threadIdx.x >> 5, lane = threadIdx.x & 31;
    int e = blockIdx.x * 8 + w;
    if (e >= nE) return;
    float t = *tptr;
    int s = src[e], d = dst[e];
    if (e + 8 < nE) {  // look-ahead prefetch of the next gather row -> global_prefetch_b8
        int s2 = src[e + 8];
        __builtin_prefetch(h + (size_t)s2 * C + lane * 4, 0, 1);
    }
    float4 v = ((const float4*)(h + (size_t)s * C))[lane];
    if (eattr) {
        float4 a = ((const float4*)(eattr + (size_t)e * C))[lane];
        v.x += a.x; v.y += a.y; v.z += a.z; v.w += a.w;
    }
    float l0 = (fmaxf(v.x, 0.0f) + GEPS) * t;
    float l1 = (fmaxf(v.y, 0.0f) + GEPS) * t;
    float l2 = (fmaxf(v.z, 0.0f) + GEPS) * t;
    float l3 = (fmaxf(v.w, 0.0f) + GEPS) * t;
    float* base = mx + (size_t)d * C + lane * 4;
    atomicMaxFloat(base + 0, l0);
    atomicMaxFloat(base + 1, l1);
    atomicMaxFloat(base + 2, l2);
    atomicMaxFloat(base + 3, l3);
}

// ---------------------------------------------------------------------------
// Pass B: alpha = exp(logit - mx[dst]); denom += alpha; aggr += alpha*msg
// ---------------------------------------------------------------------------
__global__ void scatter_sum_kernel(const float* __restrict__ h,
                                   const float* __restrict__ eattr,
                                   const int* __restrict__ src,
                                   const int* __restrict__ dst,
                                   const float* __restrict__ tptr,
                                   const float* __restrict__ mx,
                                   float* denom, float* aggr, int nE) {
    int w = threadIdx.x >> 5, lane = threadIdx.x & 31;
    int e = blockIdx.x * 8 + w;
    if (e >= nE) return;
    float t = *tptr;
    int s = src[e], d = dst[e];
    if (e + 8 < nE) {
        int s2 = src[e + 8];
        __builtin_prefetch(h + (size_t)s2 * C + lane * 4, 0, 1);
    }
    float4 v = ((const float4*)(h + (size_t)s * C))[lane];
    if (eattr) {
        float4 a = ((const float4*)(eattr + (size_t)e * C))[lane];
        v.x += a.x; v.y += a.y; v.z += a.z; v.w += a.w;
    }
    float m0 = fmaxf(v.x, 0.0f) + GEPS;
    float m1 = fmaxf(v.y, 0.0f) + GEPS;
    float m2 = fmaxf(v.z, 0.0f) + GEPS;
    float m3 = fmaxf(v.w, 0.0f) + GEPS;
    float4 mv = ((const float4*)(mx + (size_t)d * C))[lane];
    float a0 = __expf(m0 * t - mv.x);
    float a1 = __expf(m1 * t - mv.y);
    float a2 = __expf(m2 * t - mv.z);
    float a3 = __expf(m3 * t - mv.w);
    size_t base = (size_t)d * C + lane * 4;
    atomicAdd(denom + base + 0, a0);
    atomicAdd(denom + base + 1, a1);
    atomicAdd(denom + base + 2, a2);
    atomicAdd(denom + base + 3, a3);
    atomicAdd(aggr + base + 0, a0 * m0);
    atomicAdd(aggr + base + 1, a1 * m1);
    atomicAdd(aggr + base + 2, a2 * m2);
    atomicAdd(aggr + base + 3, a3 * m3);
}

// ---------------------------------------------------------------------------
// Fused: in = aggr/denom + h ; T = relu(LN(in@W1+b1)) ; y = T@W2+b2 ; out = x+y
// 32 rows/block, 256 threads = 8 waves. Each wave owns a column slab across
// BOTH 16-row M tiles, so every W1/W2 fragment is loaded from L2 exactly once
// per block and feeds 2 WMMAs (A fragments likewise reused x2 from LDS).
// out may alias aggr (rows are block-private, read fully before written).
// ---------------------------------------------------------------------------
__launch_bounds__(256)
__global__ void mlp_wmma_kernel(const float* aggr,
                                const float* __restrict__ denom,
                                const float* __restrict__ h,
                                const float* __restrict__ x_in,
                                const float* __restrict__ W1,
                                const float* __restrict__ b1,
                                const float* __restrict__ lng,
                                const float* __restrict__ lnb,
                                const float* __restrict__ W2,
                                const float* __restrict__ b2,
                                float* out, int nrows) {
    __shared__ float s_in[32][132];   // stride 132: 4-bank skew per row
    __shared__ float s_h[32][260];    // stride 260: 4-bank skew per row
    const int tid = threadIdx.x, lane = tid & 31, w = tid >> 5;
    const int row0 = blockIdx.x * 32;

    for (int i = tid; i < 32 * C; i += 256) {
        int r = i >> 7, c = i & 127;
        size_t g = (size_t)(row0 + r) * C + c;
        float dn = denom[g];
        dn = (dn == 0.0f) ? 1.0f : dn;
        s_in[r][c] = aggr[g] / dn + h[g];
    }
    __syncthreads();

    const int mrow  = lane & 15;          // M index within tile (A & B low lanes)
    const int khalf = (lane >> 4) << 1;   // K offset (0 or 2)
    const int mbase = (lane >> 4) << 3;   // C/D row offset (0 or 8)

    // ---- GEMM1: 32x128 @ 128x256 ; wave w -> columns [w*32, w*32+32) ----
    v8f acc[2][2] = {};                   // [mtile][ntile]
    for (int k0 = 0; k0 < C; k0 += 4) {
        int kk = k0 + khalf;
        v2f a0, a1;
        a0[0] = s_in[mrow][kk];      a0[1] = s_in[mrow][kk + 1];
        a1[0] = s_in[16 + mrow][kk]; a1[1] = s_in[16 + mrow][kk + 1];
        #pragma unroll
        for (int j = 0; j < 2; j++) {
            int n = w * 32 + j * 16 + mrow;
            v2f bf; bf[0] = W1[(size_t)kk * C2 + n];
                    bf[1] = W1[(size_t)(kk + 1) * C2 + n];
            acc[0][j] = WMMA_F32(a0, bf, acc[0][j]);
            acc[1][j] = WMMA_F32(a1, bf, acc[1][j]);
        }
    }
    #pragma unroll
    for (int mt = 0; mt < 2; mt++) {
        #pragma unroll
        for (int j = 0; j < 2; j++) {
            int n = w * 32 + j * 16 + mrow;
            float bb = b1[n];
            #pragma unroll
            for (int r = 0; r < 8; r++)
                s_h[mt * 16 + mbase + r][n] = acc[mt][j][r] + bb;
        }
    }
    __syncthreads();

    // ---- LayerNorm(256) + ReLU, one wave per 4 rows (8 waves x 4 = 32) ----
    for (int rr = 0; rr < 4; rr++) {
        int r = w * 4 + rr;
        float vals[8], s = 0.0f, sq = 0.0f;
        #pragma unroll
        for (int i = 0; i < 8; i++) {
            float v = s_h[r][lane * 8 + i];
            vals[i] = v; s += v; sq += v * v;
        }
        s = wave_sum(s); sq = wave_sum(sq);
        float m  = s * (1.0f / C2);
        float vr = sq * (1.0f / C2) - m * m;
        float rs = rsqrtf(vr + 1e-5f);
        #pragma unroll
        for (int i = 0; i < 8; i++) {
            int c = lane * 8 + i;
            s_h[r][c] = fmaxf((vals[i] - m) * rs * lng[c] + lnb[c], 0.0f);
        }
    }
    __syncthreads();

    // ---- GEMM2: 32x256 @ 256x128 ; wave w -> columns [w*16, w*16+16) ----
    v8f acc2[2] = {};                     // [mtile]
    for (int k0 = 0; k0 < C2; k0 += 4) {
        int kk = k0 + khalf;
        v2f a0, a1;
        a0[0] = s_h[mrow][kk];      a0[1] = s_h[mrow][kk + 1];
        a1[0] = s_h[16 + mrow][kk]; a1[1] = s_h[16 + mrow][kk + 1];
        int n = w * 16 + mrow;
        v2f bf; bf[0] = W2[(size_t)kk * C + n];
                bf[1] = W2[(size_t)(kk + 1) * C + n];
        acc2[0] = WMMA_F32(a0, bf, acc2[0]);
        acc2[1] = WMMA_F32(a1, bf, acc2[1]);
    }
    #pragma unroll
    for (int mt = 0; mt < 2; mt++) {
        int n = w * 16 + mrow;
        float bb = b2[n];
        #pragma unroll
        for (int r = 0; r < 8; r++) {
            size_t g = (size_t)(row0 + mt * 16 + mbase + r) * C + n;
            out[g] = x_in[g] + acc2[mt][r] + bb;
        }
    }
    (void)nrows;
}

extern "C" void kernel_launch(void* const* d_in, const int* in_sizes, int n_in,
                              void* d_out, int out_size, void* d_ws, size_t ws_size,
                              hipStream_t stream) {
    const float* v_x = (const float*)d_in[0];
    const float* e_x = (const float*)d_in[1];
    const int* v_ei  = (const int*)d_in[2];   // [2*E]  row0=src row1=dst
    const int* e_ei  = (const int*)d_in[3];   // [2*LE]
    const float* e_pre_g = (const float*)d_in[4];
    const float* e_pre_b = (const float*)d_in[5];
    const float* e_t     = (const float*)d_in[6];
    const float* e_W1    = (const float*)d_in[7];
    const float* e_b1    = (const float*)d_in[8];
    const float* e_ln_g  = (const float*)d_in[9];
    const float* e_ln_b  = (const float*)d_in[10];
    const float* e_W2    = (const float*)d_in[11];
    const float* e_b2    = (const float*)d_in[12];
    const float* n_pre_g = (const float*)d_in[13];
    const float* n_pre_b = (const float*)d_in[14];
    const float* n_t     = (const float*)d_in[15];
    const float* n_W1    = (const float*)d_in[16];
    const float* n_b1    = (const float*)d_in[17];
    const float* n_ln_g  = (const float*)d_in[18];
    const float* n_ln_b  = (const float*)d_in[19];
    const float* n_W2    = (const float*)d_in[20];
    const float* n_b2    = (const float*)d_in[21];
    (void)in_sizes; (void)n_in; (void)out_size; (void)ws_size;

    float* v_out = (float*)d_out;                       // [N*C]
    float* e_out = (float*)d_out + (size_t)NN * C;      // [E*C]

    float* h   = (float*)d_ws;                          // max(E,N)*C
    float* mx  = h  + (size_t)EE * C;
    float* dnm = mx + (size_t)EE * C;

    const size_t EC = (size_t)EE * C, NC = (size_t)NN * C;

    // ---------- edge layer: line graph (E nodes, LE edges, no edge_attr) ----------
    prenorm_kernel<<<EE / 8, 256, 0, stream>>>(e_x, e_pre_g, e_pre_b, h, EE);
    fill_kernel<<<2048, 256, 0, stream>>>(mx, -INFINITY, EC);
    fill_kernel<<<2048, 256, 0, stream>>>(dnm, 0.0f, EC);
    fill_kernel<<<2048, 256, 0, stream>>>(e_out, 0.0f, EC);
    scatter_max_kernel<<<LEE / 8, 256, 0, stream>>>(h, nullptr, e_ei, e_ei + LEE, e_t, mx, LEE);
    scatter_sum_kernel<<<LEE / 8, 256, 0, stream>>>(h, nullptr, e_ei, e_ei + LEE, e_t, mx, dnm, e_out, LEE);
    mlp_wmma_kernel<<<EE / 32, 256, 0, stream>>>(e_out, dnm, h, e_x,
        e_W1, e_b1, e_ln_g, e_ln_b, e_W2, e_b2, e_out, EE);

    // ---------- node layer: node graph (N nodes, E edges, edge_attr = e_out) ----------
    prenorm_kernel<<<NN / 8, 256, 0, stream>>>(v_x, n_pre_g, n_pre_b, h, NN);
    fill_kernel<<<512, 256, 0, stream>>>(mx, -INFINITY, NC);
    fill_kernel<<<512, 256, 0, stream>>>(dnm, 0.0f, NC);
    fill_kernel<<<512, 256, 0, stream>>>(v_out, 0.0f, NC);
    scatter_max_kernel<<<EE / 8, 256, 0, stream>>>(h, e_out, v_ei, v_ei + EE, n_t, mx, EE);
    scatter_sum_kernel<<<EE / 8, 256, 0, stream>>>(h, e_out, v_ei, v_ei + EE, n_t, mx, dnm, v_out, EE);
    mlp_wmma_kernel<<<NN / 32, 256, 0, stream>>>(v_out, dnm, h, v_x,
        n_W1, n_b1, n_ln_g, n_ln_b, n_W2, n_b2, v_out, NN);
}